// ifft2D_4277787427261
// MI455X (gfx1250) — compile-verified
//
#include <hip/hip_runtime.h>
#include <math.h>

// ---------------------------------------------------------------------------
// Batched 2D inverse DFT (512x512, batch 64) + magnitude, via Cooley-Tukey
// factorization F512 = (F16 (x) I32) * T * (I16 (x) F32), all stages executed
// as f32 complex WMMA GEMMs (V_WMMA_F32_16X16X4_F32, CDNA5 wave32).
//   pass 1: R1T = (W512 * C_a)^T          (complex, workspace / L2-resident)
//   pass 2: out = | (W512 * R1T)^T |      (magnitude f32, final layout)
// 10.7x fewer FLOPs than the dense DFT -> compute (~21us) matches the
// 23.3 TB/s memory floor (~20us) at full f32 matrix precision.
// ---------------------------------------------------------------------------

typedef float v2f __attribute__((ext_vector_type(2)));
typedef float v8f __attribute__((ext_vector_type(8)));

#define NN   512
#define NB   64
#define COLS 16                  // columns of X handled per workgroup
#define AINT_STRIDE 17           // padded LDS row stride (bank decorrelation)

__device__ __forceinline__ v8f wmma4(v2f a, v2f b, v8f c) {
    return __builtin_amdgcn_wmma_f32_16x16x4_f32(false, a, false, b,
                                                 (short)0, c, false, false);
}

// Dr += Ar*Br - Ai*Bi ; Di += Ar*Bi + Ai*Br   (Ain = -Ai precomputed on VALU)
__device__ __forceinline__ void cwmma(v2f Ar, v2f Ai, v2f Ain,
                                      v2f Br, v2f Bi, v8f& Dr, v8f& Di) {
    Dr = wmma4(Ar,  Br, Dr);
    Dr = wmma4(Ain, Bi, Dr);
    Di = wmma4(Ar,  Bi, Di);
    Di = wmma4(Ai,  Br, Di);
}

// Tables: W16[k1,n1]=e^{2pi i k1 n1/16}/16 ; W32[k2,n2]=e^{2pi i k2 n2/32}/32 ;
//         T[k1,n2]=e^{2pi i k1 n2/512}. Layout: [0,256) W16, [256,1280) W32,
//         [1280,1792) T.
__global__ void build_tables_kernel(float2* __restrict__ tab) {
    const int i = blockIdx.x * blockDim.x + threadIdx.x;
    const float TWO_PI = 6.28318530717958647692f;
    float s, c;
    if (i < 256) {
        int k1 = i >> 4, n1 = i & 15;
        int p = (k1 * n1) & 15;
        sincosf(TWO_PI * (float)p * (1.0f / 16.0f), &s, &c);
        float2 w; w.x = c * (1.0f / 16.0f); w.y = s * (1.0f / 16.0f);
        tab[i] = w;
    } else if (i < 1280) {
        int j = i - 256;
        int k2 = j >> 5, n2 = j & 31;
        int p = (k2 * n2) & 31;
        sincosf(TWO_PI * (float)p * (1.0f / 32.0f), &s, &c);
        float2 w; w.x = c * (1.0f / 32.0f); w.y = s * (1.0f / 32.0f);
        tab[i] = w;
    } else if (i < 1792) {
        int j = i - 1280;
        int k1 = j >> 5, n2 = j & 31;                 // k1*n2 <= 465 < 512
        sincosf(TWO_PI * (float)(k1 * n2) * (1.0f / 512.0f), &s, &c);
        float2 w; w.x = c; w.y = s;
        tab[i] = w;
    }
}

// One workgroup (8 waves): full 512-point transform of COLS columns of X,
// output written TRANSPOSED (coalesced via LDS staging).
// MAG=false: Yc[c0+c][y] = complex result. MAG=true: Ym[c0+c][y] = |result|.
template <bool MAG>
__device__ __forceinline__ void transform512(const float2* __restrict__ X,
                                             float2* __restrict__ Yc,
                                             float* __restrict__ Ym,
                                             const float2* __restrict__ W16m,
                                             const float2* __restrict__ W32m,
                                             const float2* __restrict__ Tw,
                                             int c0) {
    __shared__ float2 Aint[32 * 16 * AINT_STRIDE];   // [n2][k1][c], padded rows
    __shared__ float2 Tbuf[COLS * NN];               // [c][y] transpose staging

    const int tid  = threadIdx.x;
    const int wave = tid >> 5;
    const int lane = tid & 31;
    const int half = lane >> 4;
    const int l16  = lane & 15;
    const int k0   = 2 * half;

    // ---- stage 1: A[k1,n2,c] = T[k1,n2] * sum_n1 W16[k1,n1] X[32n1+n2, c] ----
    for (int i = 0; i < 4; ++i) {
        const int n2 = wave * 4 + i;
        v8f Dr = (v8f){0.f,0.f,0.f,0.f,0.f,0.f,0.f,0.f};
        v8f Di = (v8f){0.f,0.f,0.f,0.f,0.f,0.f,0.f,0.f};
        for (int k = 0; k < 16; k += 4) {
            const int kk = k + k0;
            const float4 a4 = *(const float4*)(&W16m[l16 * 16 + kk]);
            const v2f Ar  = { a4.x,  a4.z};
            const v2f Ai  = { a4.y,  a4.w};
            const v2f Ain = {-a4.y, -a4.w};
            const float2 b0 = X[(size_t)(32 * kk       + n2) * NN + c0 + l16];
            const float2 b1 = X[(size_t)(32 * (kk + 1) + n2) * NN + c0 + l16];
            const v2f Br = {b0.x, b1.x};
            const v2f Bi = {b0.y, b1.y};
            cwmma(Ar, Ai, Ain, Br, Bi, Dr, Di);
        }
#pragma unroll
        for (int v = 0; v < 8; ++v) {
            const int k1 = 8 * half + v;
            const float2 tw = Tw[k1 * 32 + n2];
            float2 o;
            o.x = Dr[v] * tw.x - Di[v] * tw.y;
            o.y = Dr[v] * tw.y + Di[v] * tw.x;
            Aint[(n2 * 16 + k1) * AINT_STRIDE + l16] = o;
        }
    }
    __syncthreads();

    // ---- stage 2: Y[k1+16k2, c] = sum_n2 W32[k2,n2] A[k1,n2,c] ----
    for (int i = 0; i < 2; ++i) {
        const int k1 = wave + 8 * i;
#pragma unroll
        for (int mt = 0; mt < 2; ++mt) {
            v8f Dr = (v8f){0.f,0.f,0.f,0.f,0.f,0.f,0.f,0.f};
            v8f Di = (v8f){0.f,0.f,0.f,0.f,0.f,0.f,0.f,0.f};
            for (int k = 0; k < 32; k += 4) {
                const int kk = k + k0;
                const float4 a4 = *(const float4*)(&W32m[(16 * mt + l16) * 32 + kk]);
                const v2f Ar  = { a4.x,  a4.z};
                const v2f Ai  = { a4.y,  a4.w};
                const v2f Ain = {-a4.y, -a4.w};
                const float2 b0 = Aint[((kk    ) * 16 + k1) * AINT_STRIDE + l16];
                const float2 b1 = Aint[((kk + 1) * 16 + k1) * AINT_STRIDE + l16];
                const v2f Br = {b0.x, b1.x};
                const v2f Bi = {b0.y, b1.y};
                cwmma(Ar, Ai, Ain, Br, Bi, Dr, Di);
            }
#pragma unroll
            for (int v = 0; v < 8; ++v) {
                const int k2 = 16 * mt + 8 * half + v;
                const int y  = k1 + 16 * k2;        // output index, 0..511
                if (MAG) {
                    ((float*)Tbuf)[l16 * NN + y] =
                        sqrtf(Dr[v] * Dr[v] + Di[v] * Di[v]);
                } else {
                    float2 o; o.x = Dr[v]; o.y = Di[v];
                    Tbuf[l16 * NN + y] = o;
                }
            }
        }
    }
    __syncthreads();

    // ---- coalesced transposed writeback: rows (c0+c) of length 512 ----
    if (MAG) {
        const float* Tf = (const float*)Tbuf;
        for (int i = tid; i < COLS * NN; i += 256) {
            const int c = i >> 9, y = i & (NN - 1);
            Ym[(size_t)(c0 + c) * NN + y] = Tf[c * NN + y];
        }
    } else {
        for (int i = tid; i < COLS * NN; i += 256) {
            const int c = i >> 9, y = i & (NN - 1);
            Yc[(size_t)(c0 + c) * NN + y] = Tbuf[c * NN + y];
        }
    }
}

__global__ __launch_bounds__(256) void pass1_kernel(const float2* __restrict__ Cin,
                                                    float2* __restrict__ R1T,
                                                    const float2* __restrict__ tab,
                                                    int batch0) {
    const int a = batch0 + blockIdx.y;
    transform512<false>(Cin + (size_t)a * NN * NN,
                        R1T + (size_t)blockIdx.y * NN * NN, nullptr,
                        tab, tab + 256, tab + 1280, blockIdx.x * COLS);
}

__global__ __launch_bounds__(256) void pass2_kernel(const float2* __restrict__ R1T,
                                                    float* __restrict__ Out,
                                                    const float2* __restrict__ tab,
                                                    int batch0) {
    transform512<true>(R1T + (size_t)blockIdx.y * NN * NN, nullptr,
                       Out + (size_t)(batch0 + blockIdx.y) * NN * NN,
                       tab, tab + 256, tab + 1280, blockIdx.x * COLS);
}

extern "C" void kernel_launch(void* const* d_in, const int* in_sizes, int n_in,
                              void* d_out, int out_size, void* d_ws, size_t ws_size,
                              hipStream_t stream) {
    (void)in_sizes; (void)n_in; (void)out_size;

    const float2* inp = (const float2*)d_in[0];   // [64,512,512] interleaved complex
    float* out        = (float*)d_out;            // [64,512,512,1] magnitude

    // Workspace: tables (1792 float2, padded to 16 KB), then R1T chunk slots.
    const size_t tabBytes = 16384;
    const size_t perBatch = (size_t)NN * NN * sizeof(float2);   // 2 MB
    float2* tab = (float2*)d_ws;
    float2* R1T = (float2*)((char*)d_ws + tabBytes);

    int G = NB;
    if (ws_size > tabBytes) {
        size_t avail = (ws_size - tabBytes) / perBatch;
        if (avail < 1) avail = 1;
        if (avail < (size_t)G) G = (int)avail;
    } else {
        G = 1;
    }

    build_tables_kernel<<<7, 256, 0, stream>>>(tab);

    for (int b0 = 0; b0 < NB; b0 += G) {
        const int g = (NB - b0) < G ? (NB - b0) : G;
        dim3 grid(NN / COLS, g);
        pass1_kernel<<<grid, 256, 0, stream>>>(inp, R1T, tab, b0);
        pass2_kernel<<<grid, 256, 0, stream>>>(R1T, out, tab, b0);
    }
}